// FNO_45019847197407
// MI455X (gfx1250) — compile-verified
//
#include <hip/hip_runtime.h>
#include <math.h>

typedef __attribute__((ext_vector_type(2))) float v2f;
typedef __attribute__((ext_vector_type(8))) float v8f;
typedef __attribute__((__vector_size__(4 * sizeof(int)))) int i4vec;

#define NB 8
#define NC 32
#define NH 256
#define NWD 256
#define HW (NH*NWD)
#define PIX (NB*HW)

#if defined(__has_builtin)
#if __has_builtin(__builtin_amdgcn_global_load_async_to_lds_b128) && \
    __has_builtin(__builtin_amdgcn_s_wait_asynccnt)
#define HAVE_ASYNC_LDS 1
#endif
#endif
#ifndef HAVE_ASYNC_LDS
#define HAVE_ASYNC_LDS 0
#endif

// Async global->LDS 16B copy (CDNA5 GLOBAL_LOAD_ASYNC_TO_LDS_B128, ASYNCcnt),
// with a plain float4 fallback if the builtin is not declared.
static __device__ __forceinline__ void async_copy_b128(float4* lds_dst, const float4* gsrc) {
#if HAVE_ASYNC_LDS
  __builtin_amdgcn_global_load_async_to_lds_b128((i4vec*)gsrc, (i4vec*)lds_dst, 0, 0);
#else
  *lds_dst = *gsrc;
#endif
}

static __device__ __forceinline__ void async_drain() {
#if HAVE_ASYNC_LDS
  __builtin_amdgcn_s_wait_asynccnt(0);
#endif
}

static __device__ __forceinline__ v8f wmma4(v2f a, v2f b, v8f c) {
  // D = A(16x4 f32) * B(4x16 f32) + C(16x16 f32)
  return __builtin_amdgcn_wmma_f32_16x16x4_f32(false, a, false, b, (short)0, c, false, false);
}

static __device__ __forceinline__ float gelu_exact(float x) {
  return 0.5f * x * (1.0f + erff(x * 0.70710678118654752f));
}

// ---------------------------------------------------------------------------
// K0: build DFT tables (exact phase via integer mod 256)
//   FwT[slot][w] slot<16: cos(2pi w kx/256), slot>=16: -sin  (forward W-DFT, B^T)
//   THre/THim[ky'][h]: forward H-DFT twiddles (ky = ky'<16 ? ky' : 224+ky')
//   IWc/IWs[w][kx]: inverse-real W-DFT incl. 2x symmetry factor and 1/(H*W)
// ---------------------------------------------------------------------------
__global__ void k_tables(float* __restrict__ FwT, float* __restrict__ THre,
                         float* __restrict__ THim, float* __restrict__ IWc,
                         float* __restrict__ IWs) {
  int t = blockIdx.x * blockDim.x + threadIdx.x;   // 8192 threads
  const float step = 6.28318530717958647692f / 256.0f;
  if (t < 32 * 256) {
    int slot = t >> 8, w = t & 255;
    int kx = slot & 15;
    float th = step * (float)((w * kx) & 255);
    FwT[t] = (slot < 16) ? cosf(th) : -sinf(th);
    int ky = (slot < 16) ? slot : (224 + slot);
    float th2 = step * (float)((w * ky) & 255);   // w plays role of h here
    THre[t] = cosf(th2);
    THim[t] = -sinf(th2);
  }
  if (t < 256 * 16) {
    int w = t >> 4, kx = t & 15;
    float th = step * (float)((w * kx) & 255);
    float c = ((kx == 0) ? 1.0f : 2.0f) * (1.0f / 65536.0f);
    IWc[t] = c * cosf(th);
    IWs[t] = -c * sinf(th);
  }
}

// ---------------------------------------------------------------------------
// K1: lift  [B,H,W,10] + grid -> [B,C,H,W]
// ---------------------------------------------------------------------------
__global__ void __launch_bounds__(256) k_lift(const float* __restrict__ x,
                                              const float* __restrict__ fc0_w,
                                              const float* __restrict__ fc0_b,
                                              float* __restrict__ out) {
  __shared__ float wS[12 * 32];
  __shared__ float bS[32];
  int tid = threadIdx.x;
  for (int i = tid; i < 12 * 32; i += 256) wS[i] = fc0_w[i];
  if (tid < 32) bS[tid] = fc0_b[tid];
  __syncthreads();
  int p = blockIdx.x * 256 + tid;
  int b = p >> 16;
  int rem = p & 65535;
  int hh = rem >> 8, ww = rem & 255;
  float in[12];
  const float* xp = x + (size_t)p * 10;
  #pragma unroll
  for (int i = 0; i < 10; i++) in[i] = xp[i];
  in[10] = (float)hh * (1.0f / 255.0f);
  in[11] = (float)ww * (1.0f / 255.0f);
  float acc[32];
  #pragma unroll
  for (int c = 0; c < 32; c++) acc[c] = bS[c];
  #pragma unroll
  for (int t = 0; t < 12; t++) {
    float v = in[t];
    #pragma unroll
    for (int c = 0; c < 32; c++) acc[c] = fmaf(v, wS[t * 32 + c], acc[c]);
  }
  size_t base = (size_t)b * 32 * HW + (size_t)hh * 256 + ww;
  #pragma unroll
  for (int c = 0; c < 32; c++) out[base + (size_t)c * HW] = acc[c];
}

// ---------------------------------------------------------------------------
// K2: forward W-DFT as WMMA GEMM: [R=65536 x 256] * [256 x 32] -> Xw[R][32]
//     (cols 0..15 = Re X[kx], cols 16..31 = Im X[kx])
//     block = 128 threads (4 waves), 32-row tile, full K in LDS
//     A/B tiles staged with GLOBAL_LOAD_ASYNC_TO_LDS_B128
// ---------------------------------------------------------------------------
__global__ void __launch_bounds__(128) k_fwdW(const float* __restrict__ src,
                                              const float* __restrict__ FwT,
                                              float* __restrict__ Xw) {
  __shared__ float As[32 * 256];
  __shared__ float Bt[32 * 256];
  int tid = threadIdx.x;
  const float4* g = (const float4*)(src + (size_t)blockIdx.x * 32 * 256);
  const float4* f4 = (const float4*)FwT;
  float4* a4 = (float4*)As;
  float4* b4 = (float4*)Bt;
  for (int i = tid; i < 2048; i += 128) {
    __builtin_prefetch(g + i + 2048, 0, 1);   // next block's rows -> global_prefetch
    async_copy_b128(a4 + i, g + i);
    async_copy_b128(b4 + i, f4 + i);
  }
  async_drain();
  __syncthreads();
  int wid = tid >> 5, lane = tid & 31;
  int lo = lane & 15, hi = lane >> 4;
  int wy = wid >> 1, wx = wid & 1;
  const float* Ap = As + (16 * wy + lo) * 256 + 2 * hi;
  const float* Bp = Bt + (16 * wx + lo) * 256 + 2 * hi;
  v8f acc = {};
  #pragma unroll 8
  for (int k0 = 0; k0 < 256; k0 += 4) {
    v2f a = *(const v2f*)(Ap + k0);
    v2f b = *(const v2f*)(Bp + k0);
    acc = wmma4(a, b, acc);
  }
  size_t rbase = (size_t)blockIdx.x * 32 + 16 * wy + 8 * hi;
  int col = 16 * wx + lo;
  #pragma unroll
  for (int j = 0; j < 8; j++) Xw[(rbase + j) * 32 + col] = acc[j];
}

// ---------------------------------------------------------------------------
// K3: forward H-DFT (32 kept freqs) per (b,c). Out layout [b][ky'][kx][c]
// ---------------------------------------------------------------------------
__global__ void __launch_bounds__(512) k_fwdH(const float* __restrict__ Xw,
                                              const float* __restrict__ THre,
                                              const float* __restrict__ THim,
                                              float* __restrict__ Xfr,
                                              float* __restrict__ Xfi) {
  __shared__ float S[256 * 32];
  int tid = threadIdx.x;
  int bc = blockIdx.x;                       // b*32+c
  const float4* g = (const float4*)(Xw + (size_t)bc * 256 * 32);
  float4* s4 = (float4*)S;
  for (int i = tid; i < 2048; i += 512) async_copy_b128(s4 + i, g + i);
  async_drain();
  __syncthreads();
  int kyp = tid >> 4, kx = tid & 15;
  const float* tr = THre + kyp * 256;
  const float* ti = THim + kyp * 256;
  float re = 0.f, im = 0.f;
  #pragma unroll 8
  for (int h = 0; h < 256; h++) {
    float sr = S[h * 32 + kx], si = S[h * 32 + kx + 16];
    float cr = tr[h], ci = ti[h];
    re += sr * cr - si * ci;
    im += sr * ci + si * cr;
  }
  int b = bc >> 5, c = bc & 31;
  size_t o = (((size_t)b * 32 + kyp) * 16 + kx) * 32 + c;
  Xfr[o] = re;
  Xfi[o] = im;
}

// ---------------------------------------------------------------------------
// K4: complex mode multiply: Yf[b,o,ky',kx] = sum_c Xf[b,c,ky',kx]*W[c,o,ky,kx]
// ---------------------------------------------------------------------------
__global__ void __launch_bounds__(256) k_modemul(const float* __restrict__ Xfr,
                                                 const float* __restrict__ Xfi,
                                                 const float* __restrict__ w1,
                                                 const float* __restrict__ w2,
                                                 float* __restrict__ Yfr,
                                                 float* __restrict__ Yfi) {
  int t = blockIdx.x * 256 + threadIdx.x;    // 131072 total
  int o = t & 31;
  int kx = (t >> 5) & 15;
  int kyp = (t >> 9) & 31;
  int b = t >> 14;
  const float* wbase = (kyp < 16) ? w1 : w2;
  int xm = kyp & 15;
  size_t widx = ((size_t)o * 16 + xm) * 16 + kx;   // + c*8192 per input channel
  const float* xr = Xfr + (((size_t)b * 32 + kyp) * 16 + kx) * 32;
  const float* xi = Xfi + (((size_t)b * 32 + kyp) * 16 + kx) * 32;
  float yr = 0.f, yi = 0.f;
  #pragma unroll 4
  for (int c = 0; c < 32; c++) {
    size_t wi2 = (widx + (size_t)c * 8192) * 2;
    float wr = wbase[wi2], wim = wbase[wi2 + 1];
    float ar = xr[c], ai = xi[c];
    yr += ar * wr - ai * wim;
    yi += ar * wim + ai * wr;
  }
  size_t oidx = (((size_t)b * 32 + o) * 32 + kyp) * 16 + kx;
  Yfr[oidx] = yr;
  Yfi[oidx] = yi;
}

// ---------------------------------------------------------------------------
// K5: inverse H-DFT from 32 modes: Yh[b,o,h,kx] = sum_ky' Yf * conj(TH)
// ---------------------------------------------------------------------------
__global__ void __launch_bounds__(256) k_invH(const float* __restrict__ Yfr,
                                              const float* __restrict__ Yfi,
                                              const float* __restrict__ THre,
                                              const float* __restrict__ THim,
                                              float* __restrict__ Yhr,
                                              float* __restrict__ Yhi) {
  int t = blockIdx.x * 256 + threadIdx.x;    // 1048576 total
  int kx = t & 15;
  int h = (t >> 4) & 255;
  int o = (t >> 12) & 31;
  int b = t >> 17;
  const float* yr0 = Yfr + (((size_t)b * 32 + o) * 32) * 16 + kx;
  const float* yi0 = Yfi + (((size_t)b * 32 + o) * 32) * 16 + kx;
  float re = 0.f, im = 0.f;
  #pragma unroll 8
  for (int k = 0; k < 32; k++) {
    float yr = yr0[k * 16], yi = yi0[k * 16];
    float cr = THre[k * 256 + h];
    float ci = -THim[k * 256 + h];           // conj => e^{+i theta}
    re += yr * cr - yi * ci;
    im += yr * ci + yi * cr;
  }
  size_t oidx = (((size_t)b * 32 + o) * 256 + h) * 16 + kx;
  Yhr[oidx] = re;
  Yhi[oidx] = im;
}

// ---------------------------------------------------------------------------
// K6: fused inverse W-DFT (x1) + 1x1 conv (x2, WMMA) + bias + GELU + skip
//     one block per (b, h-row); 256 threads = 8 waves, 32 tile-jobs (2x16)
// ---------------------------------------------------------------------------
__global__ void __launch_bounds__(256) k_spec_conv(
    const float* __restrict__ hin,
    const float* __restrict__ Yhr, const float* __restrict__ Yhi,
    const float* __restrict__ IWc, const float* __restrict__ IWs,
    const float* __restrict__ cw, const float* __restrict__ cb,
    const float* __restrict__ skip, int add_skip,
    float* __restrict__ hout) {
  __shared__ float hT[256 * 34];       // [w][c], padded for bank/align
  __shared__ float CW[32 * 32];        // [o][c]
  __shared__ float cbS[32];
  __shared__ float YhrS[32 * 16];      // [o][kx]
  __shared__ float YhiS[32 * 16];
  int tid = threadIdx.x;
  int b = blockIdx.x >> 8;
  int hrow = blockIdx.x & 255;
  {
    const float* src = hin + (size_t)b * 32 * HW + (size_t)hrow * 256;
    #pragma unroll 4
    for (int c = 0; c < 32; c++) hT[tid * 34 + c] = src[(size_t)c * HW + tid];
  }
  for (int i = tid; i < 1024; i += 256) CW[i] = cw[i];
  if (tid < 32) cbS[tid] = cb[tid];
  {
    size_t base = ((size_t)b * 32 * 256 + hrow) * 16;
    for (int i = tid; i < 512; i += 256) {
      int o = i >> 4, kx = i & 15;
      size_t gi = base + (size_t)o * 4096 + kx;
      YhrS[i] = Yhr[gi];
      YhiS[i] = Yhi[gi];
    }
  }
  __syncthreads();
  int wid = tid >> 5, lane = tid & 31, lo = lane & 15, hi = lane >> 4;
  #pragma unroll
  for (int jj = 0; jj < 4; jj++) {
    int job = wid + 8 * jj;            // 0..31
    int nt = job & 15, mt = job >> 4;
    const float* Ap = CW + (16 * mt + lo) * 32 + 2 * hi;
    const float* Bp = hT + (16 * nt + lo) * 34 + 2 * hi;
    v8f acc = {};
    #pragma unroll
    for (int k0 = 0; k0 < 32; k0 += 4) {
      v2f a = *(const v2f*)(Ap + k0);
      v2f bb = *(const v2f*)(Bp + k0);
      acc = wmma4(a, bb, acc);
    }
    int w = 16 * nt + lo;
    float iwc[16], iws[16];
    {
      const float4* c4 = (const float4*)(IWc + w * 16);
      const float4* s4 = (const float4*)(IWs + w * 16);
      #pragma unroll
      for (int q = 0; q < 4; q++) {
        float4 cv = c4[q], sv = s4[q];
        iwc[4 * q] = cv.x; iwc[4 * q + 1] = cv.y; iwc[4 * q + 2] = cv.z; iwc[4 * q + 3] = cv.w;
        iws[4 * q] = sv.x; iws[4 * q + 1] = sv.y; iws[4 * q + 2] = sv.z; iws[4 * q + 3] = sv.w;
      }
    }
    #pragma unroll
    for (int j = 0; j < 8; j++) {
      int o = 16 * mt + 8 * hi + j;
      const float* yr = YhrS + o * 16;
      const float* yi = YhiS + o * 16;
      float x1 = 0.f;
      #pragma unroll
      for (int kx = 0; kx < 16; kx++) x1 += yr[kx] * iwc[kx] + yi[kx] * iws[kx];
      float v = gelu_exact(x1 + acc[j] + cbS[o]);
      size_t oi = (((size_t)b * 32 + o) * 256 + hrow) * 256 + w;
      if (add_skip) v += skip[oi];
      hout[oi] = v;
    }
  }
}

// ---------------------------------------------------------------------------
// K7: fused final MLP: fc1 (WMMA 16px x 128, K=32) + GELU -> LDS -> fc2 (N=5)
// ---------------------------------------------------------------------------
__global__ void __launch_bounds__(256) k_final(
    const float* __restrict__ hin,
    const float* __restrict__ fc1_w, const float* __restrict__ fc1_b,
    const float* __restrict__ fc2_w, const float* __restrict__ fc2_b,
    float* __restrict__ out) {
  __shared__ float A[16 * 36];         // [pixel][c]
  __shared__ float W1T[128 * 34];      // [n][c]
  __shared__ float b1S[128];
  __shared__ float Hd[16 * 132];       // [pixel][n]
  __shared__ float W2[128 * 5];
  __shared__ float b2S[5];
  int tid = threadIdx.x;
  int p0 = blockIdx.x * 16;
  for (int e = tid; e < 512; e += 256) {
    int pl = e >> 5, c = e & 31;
    int p = p0 + pl;
    int b = p >> 16, rem = p & 65535;
    A[pl * 36 + c] = hin[((size_t)b * 32 + c) * HW + rem];
  }
  for (int e = tid; e < 4096; e += 256) {
    int c = e >> 7, n = e & 127;
    W1T[n * 34 + c] = fc1_w[e];
  }
  if (tid < 128) b1S[tid] = fc1_b[tid];
  for (int e = tid; e < 640; e += 256) W2[e] = fc2_w[e];
  if (tid < 5) b2S[tid] = fc2_b[tid];
  __syncthreads();
  int wid = tid >> 5, lane = tid & 31, lo = lane & 15, hi = lane >> 4;
  {
    const float* Ap = A + lo * 36 + 2 * hi;
    const float* Bp = W1T + (wid * 16 + lo) * 34 + 2 * hi;
    v8f acc = {};
    #pragma unroll
    for (int k0 = 0; k0 < 32; k0 += 4) {
      v2f a = *(const v2f*)(Ap + k0);
      v2f bb = *(const v2f*)(Bp + k0);
      acc = wmma4(a, bb, acc);
    }
    int n = wid * 16 + lo;
    float bias = b1S[n];
    #pragma unroll
    for (int j = 0; j < 8; j++) {
      int pl = 8 * hi + j;
      Hd[pl * 132 + n] = gelu_exact(acc[j] + bias);
    }
  }
  __syncthreads();
  if (tid < 80) {
    int pl = tid / 5, d = tid % 5;
    float s = b2S[d];
    const float* hp = Hd + pl * 132;
    #pragma unroll 8
    for (int k = 0; k < 128; k++) s = fmaf(hp[k], W2[k * 5 + d], s);
    out[(size_t)(p0 + pl) * 5 + d] = s;
  }
}

// ---------------------------------------------------------------------------
extern "C" void kernel_launch(void* const* d_in, const int* in_sizes, int n_in,
                              void* d_out, int out_size, void* d_ws, size_t ws_size,
                              hipStream_t stream) {
  (void)in_sizes; (void)n_in; (void)out_size; (void)ws_size;
  const float* x      = (const float*)d_in[0];
  const float* fc0_w  = (const float*)d_in[1];
  const float* fc0_b  = (const float*)d_in[2];
  const float* w1     = (const float*)d_in[3];
  const float* w2     = (const float*)d_in[4];
  const float* conv_w = (const float*)d_in[5];
  const float* conv_b = (const float*)d_in[6];
  const float* fc1_w  = (const float*)d_in[7];
  const float* fc1_b  = (const float*)d_in[8];
  const float* fc2_w  = (const float*)d_in[9];
  const float* fc2_b  = (const float*)d_in[10];
  float* out = (float*)d_out;

  float* W = (float*)d_ws;
  size_t off = 0;
  auto alloc = [&](size_t n) {
    float* p = W + off;
    off += (n + 255) & ~(size_t)255;
    return p;
  };
  float* bufA = alloc(16777216);
  float* bufB = alloc(16777216);
  float* bufS = alloc(16777216);
  float* Xw   = alloc(2097152);
  float* Xfr  = alloc(131072);
  float* Xfi  = alloc(131072);
  float* Yfr  = alloc(131072);
  float* Yfi  = alloc(131072);
  float* Yhr  = alloc(1048576);
  float* Yhi  = alloc(1048576);
  float* FwT  = alloc(8192);
  float* THre = alloc(8192);
  float* THim = alloc(8192);
  float* IWc  = alloc(4096);
  float* IWs  = alloc(4096);

  k_tables<<<32, 256, 0, stream>>>(FwT, THre, THim, IWc, IWs);
  k_lift<<<PIX / 256, 256, 0, stream>>>(x, fc0_w, fc0_b, bufA);

  const size_t WSLICE = 32 * 32 * 16 * 16 * 2;   // per-layer complex weight slice
  const float* hin = bufA;
  float* outs[6] = { bufS, bufA, bufB, bufS, bufA, bufB };
  int addsk[6]   = { 0, 0, 1, 0, 0, 1 };
  for (int l = 0; l < 6; l++) {
    k_fwdW<<<2048, 128, 0, stream>>>(hin, FwT, Xw);
    k_fwdH<<<256, 512, 0, stream>>>(Xw, THre, THim, Xfr, Xfi);
    k_modemul<<<512, 256, 0, stream>>>(Xfr, Xfi, w1 + l * WSLICE, w2 + l * WSLICE, Yfr, Yfi);
    k_invH<<<4096, 256, 0, stream>>>(Yfr, Yfi, THre, THim, Yhr, Yhi);
    k_spec_conv<<<2048, 256, 0, stream>>>(hin, Yhr, Yhi, IWc, IWs,
        conv_w + l * 1024, conv_b + l * 32, bufS, addsk[l], outs[l]);
    hin = outs[l];
  }
  k_final<<<PIX / 16, 256, 0, stream>>>(bufB, fc1_w, fc1_b, fc2_w, fc2_b, out);
}